// SMoE_14302241095987
// MI455X (gfx1250) — compile-verified
//
#include <hip/hip_runtime.h>

#define B_TOK 16384
#define NEXP  64
#define DIN   128
#define DOUT  128

typedef __attribute__((ext_vector_type(2))) float v2f;
typedef __attribute__((ext_vector_type(8))) float v8f;

// ---------------------------------------------------------------------------
// Kernel 1: routing. One thread per token. Centroids staged in LDS (32KB).
// Computes sum((x-c)^2) for all 64 experts, scales by d_scale^2/128,
// picks top-3 smallest (stable tie order = lowest index, matching top_k),
// and computes weights from distances to experts 0,1,2 (reference quirk).
// ---------------------------------------------------------------------------
__global__ __launch_bounds__(256) void routing_kernel(
    const float* __restrict__ x, const float* __restrict__ d_scale,
    const float* __restrict__ centroids,
    int* __restrict__ picked, float* __restrict__ wts, int* __restrict__ counts)
{
    __shared__ float cs[NEXP * DIN];
    const int tid = threadIdx.x;
    for (int i = tid; i < NEXP * DIN; i += 256) cs[i] = centroids[i];
    __syncthreads();

    const int b = blockIdx.x * 256 + tid;
    const float* xr = x + (size_t)b * DIN;

    float acc[NEXP];
#pragma unroll
    for (int e = 0; e < NEXP; ++e) acc[e] = 0.f;

#pragma unroll 1
    for (int k = 0; k < DIN; ++k) {
        const float xv = xr[k];
#pragma unroll
        for (int e = 0; e < NEXP; ++e) {
            const float d = xv - cs[e * DIN + k];
            acc[e] = fmaf(d, d, acc[e]);
        }
    }

    const float ds = d_scale[0];
    const float scale = ds * ds * (1.0f / (float)DIN);

    float v1 = 3.4e38f, v2 = 3.4e38f, v3 = 3.4e38f;
    int   i1 = 0, i2 = 0, i3 = 0;
#pragma unroll
    for (int e = 0; e < NEXP; ++e) {
        const float d = acc[e] * scale;
        if (d < v1)      { v3 = v2; i3 = i2; v2 = v1; i2 = i1; v1 = d; i1 = e; }
        else if (d < v2) { v3 = v2; i3 = i2; v2 = d;  i2 = e; }
        else if (d < v3) { v3 = d;  i3 = e; }
    }

    // Reference quirk: weights from distance to experts 0,1,2 (not picked).
    float w0 = 1.f / (1.f + acc[0] * scale);
    float w1 = 1.f / (1.f + acc[1] * scale);
    float w2 = 1.f / (1.f + acc[2] * scale);
    const float inv = 1.f / (w0 + w1 + w2);
    w0 *= inv; w1 *= inv; w2 *= inv;

    picked[b * 3 + 0] = i1; picked[b * 3 + 1] = i2; picked[b * 3 + 2] = i3;
    wts[b * 3 + 0] = w0;    wts[b * 3 + 1] = w1;    wts[b * 3 + 2] = w2;

    atomicAdd(&counts[i1], 1);
    atomicAdd(&counts[i2], 1);
    atomicAdd(&counts[i3], 1);
}

// ---------------------------------------------------------------------------
// Kernel 2: tiny exclusive prefix sum over 64 expert counts.
// ---------------------------------------------------------------------------
__global__ void scan_kernel(const int* __restrict__ counts, int* __restrict__ offsets)
{
    if (threadIdx.x == 0) {
        int s = 0;
        for (int e = 0; e < NEXP; ++e) { offsets[e] = s; s += counts[e]; }
        offsets[NEXP] = s;
    }
}

// ---------------------------------------------------------------------------
// Kernel 3: scatter (token, weight) into per-expert buckets.
// ---------------------------------------------------------------------------
__global__ __launch_bounds__(256) void scatter_kernel(
    const int* __restrict__ picked, const float* __restrict__ wts,
    const int* __restrict__ offsets, int* __restrict__ cursors,
    int* __restrict__ btok, float* __restrict__ bw)
{
    const int b = blockIdx.x * 256 + threadIdx.x;
#pragma unroll
    for (int j = 0; j < 3; ++j) {
        const int e   = picked[b * 3 + j];
        const int pos = atomicAdd(&cursors[e], 1);
        const int idx = offsets[e] + pos;
        btok[idx] = b;
        bw[idx]   = wts[b * 3 + j];
    }
}

// ---------------------------------------------------------------------------
// Kernel 4: initialize out[b][:] = sum_j w_j * bias[e_j][:] so the expert
// GEMM can pure-accumulate via float atomics.
// ---------------------------------------------------------------------------
__global__ __launch_bounds__(256) void bias_init_kernel(
    const int* __restrict__ picked, const float* __restrict__ wts,
    const float* __restrict__ bias, float* __restrict__ out)
{
    const int idx = blockIdx.x * 256 + threadIdx.x;
    const int b = idx >> 7;
    const int o = idx & (DOUT - 1);
    float s = 0.f;
#pragma unroll
    for (int j = 0; j < 3; ++j)
        s = fmaf(wts[b * 3 + j], bias[picked[b * 3 + j] * DOUT + o], s);
    out[idx] = s;
}

// ---------------------------------------------------------------------------
// Kernel 5: per-expert GEMM with V_WMMA_F32_16X16X4_F32.
// grid = (expert e, 32-col tile ct). Block = 256 threads = 8 waves.
// W[e][:, 32ct : 32ct+32] staged in LDS as TWO stride-16 tiles (16 KB total;
// stride-16 keeps the 4x16 B-fragment reads bank-conflict-free: half-waves
// hit banks m and 32+m). Each wave computes a 16(row)x32(col) output tile:
// one A fragment feeds two WMMA chains (c0, c1) -> 2 WMMAs per 8-byte A load
// and two independent accumulation chains to hide WMMA latency.
//   16x4 f32 A layout: lane (g=lane>>4, m=lane&15): v0 = A[m][2g], v1 = A[m][2g+1]
//   4x16 f32 B layout: lane (g, n=lane&15):        v0 = B[2g][n], v1 = B[2g+1][n]
//   16x16 f32 C/D:     lane (g, n): VGPR v holds row (8g+v), col n
// ---------------------------------------------------------------------------
__global__ __launch_bounds__(256) void expert_gemm_kernel(
    const float* __restrict__ x, const float* __restrict__ W,
    const int* __restrict__ btok, const float* __restrict__ bw,
    const int* __restrict__ offsets, float* __restrict__ out)
{
    const int e  = blockIdx.x;
    const int ct = blockIdx.y;              // output columns ct*32 .. ct*32+31

    __shared__ float Wt[2 * DIN * 16];      // Wt[u][row][col], u = col half
    const int tid = threadIdx.x;
    const float* We = W + (size_t)e * DIN * DOUT + ct * 32;
    for (int i = tid; i < 2 * DIN * 16; i += 256) {
        const int u   = i >> 11;            // which 16-col half
        const int rem = i & 2047;
        const int row = rem >> 4, col = rem & 15;
        Wt[i] = We[row * DOUT + u * 16 + col];
    }
    __syncthreads();

    const int start = offsets[e];
    const int end   = offsets[e + 1];
    const int n     = end - start;

    const int wave = tid >> 5;
    const int lane = tid & 31;
    const int g = lane >> 4;                // lane half: 0 or 1
    const int m = lane & 15;                // row (A) / col (B,C)

    const float* Wt0 = Wt;                  // cols ct*32 +  0..15
    const float* Wt1 = Wt + DIN * 16;       // cols ct*32 + 16..31

    for (int t = wave; t * 16 < n; t += 8) {
        const int rowbase = start + t * 16;
        const int arow = rowbase + m;
        const int tok  = (arow < end) ? btok[arow] : btok[start];
        const float* xr = x + (size_t)tok * DIN + 2 * g;

        v8f c0 = {};
        v8f c1 = {};
#pragma unroll
        for (int kk = 0; kk < 32; ++kk) {
            const float2 av = *(const float2*)(xr + 4 * kk);
            v2f a;  a.x = av.x;  a.y = av.y;
            v2f b0, b1;
            const int r0 = (4 * kk + 2 * g) * 16 + m;
            b0.x = Wt0[r0];       b0.y = Wt0[r0 + 16];
            b1.x = Wt1[r0];       b1.y = Wt1[r0 + 16];
            c0 = __builtin_amdgcn_wmma_f32_16x16x4_f32(
                     false, a, false, b0, (short)0, c0, false, false);
            c1 = __builtin_amdgcn_wmma_f32_16x16x4_f32(
                     false, a, false, b1, (short)0, c1, false, false);
        }

        const int colbase = ct * 32;
#pragma unroll
        for (int v = 0; v < 8; ++v) {
            const int r = rowbase + g * 8 + v;   // D row -> bucket entry
            if (r < end) {
                const int   tokv = btok[r];
                const float wv   = bw[r];
                float* orow = out + (size_t)tokv * DOUT + colbase + m;
                atomicAdd(orow,      c0[v] * wv);
                atomicAdd(orow + 16, c1[v] * wv);
            }
        }
    }
}

// ---------------------------------------------------------------------------
// Workspace layout (bytes):
//   [0)        picked   : B*3 int      (196608)
//   [196608)   wts      : B*3 float    (196608)
//   [393216)   counts   : 64  int
//   [393472)   offsets  : 65  int
//   [393728)   cursors  : 64  int
//   [394240)   btok     : B*3 int      (196608)
//   [590848)   bw       : B*3 float    (196608)
//   total ~ 787456 bytes
// ---------------------------------------------------------------------------
extern "C" void kernel_launch(void* const* d_in, const int* in_sizes, int n_in,
                              void* d_out, int out_size, void* d_ws, size_t ws_size,
                              hipStream_t stream)
{
    const float* x         = (const float*)d_in[0];
    const float* d_scale   = (const float*)d_in[1];
    const float* centroids = (const float*)d_in[2];
    const float* W         = (const float*)d_in[3];
    const float* bias      = (const float*)d_in[4];
    float* out = (float*)d_out;

    char* ws = (char*)d_ws;
    int*   picked  = (int*)  (ws + 0);
    float* wts     = (float*)(ws + 196608);
    int*   counts  = (int*)  (ws + 393216);
    int*   offsets = (int*)  (ws + 393472);
    int*   cursors = (int*)  (ws + 393728);
    int*   btok    = (int*)  (ws + 394240);
    float* bw      = (float*)(ws + 590848);

    // Zero counts/offsets/cursors region (1 KB) — graph-capture safe.
    hipMemsetAsync(counts, 0, 1024, stream);

    routing_kernel<<<B_TOK / 256, 256, 0, stream>>>(x, d_scale, centroids,
                                                    picked, wts, counts);
    scan_kernel<<<1, 64, 0, stream>>>(counts, offsets);
    scatter_kernel<<<B_TOK / 256, 256, 0, stream>>>(picked, wts, offsets,
                                                    cursors, btok, bw);
    bias_init_kernel<<<(B_TOK * DOUT) / 256, 256, 0, stream>>>(picked, wts,
                                                               bias, out);
    expert_gemm_kernel<<<dim3(NEXP, DOUT / 32), 256, 0, stream>>>(
        x, W, btok, bw, offsets, out);
}